// multiHeadAttention_47347719471165
// MI455X (gfx1250) — compile-verified
//
#include <hip/hip_runtime.h>
#include <hip/hip_bf16.h>

// Problem constants (from reference): b=2, f=16, p=32 -> N=512, F=VIS=1024
#define BB 2
#define NN 512
#define FF 1024
#define GG 16
#define HD 64
#define GEO 64

// GEMM blocking
#define BM 64     // block rows
#define BN 128    // block cols (4 waves x 32)
#define KC 16     // K chunk staged in LDS (4 WMMA K-steps)
#define APAD 80   // padded A rows  (bank-shift for upper lane half)
#define BPAD 144  // padded B cols  (bank-shift for upper lane half)

typedef __attribute__((ext_vector_type(2))) float v2f;
typedef __attribute__((ext_vector_type(8))) float v8f;

// ---------------------------------------------------------------------------
// fp32 WMMA GEMM:  C[M,N] = scale * (A[M,K] x B[K,N]) (+ bias[col])
//   - 128-thread block = 4 waves; block tile 64x128, wave tile 64x32 (4x2
//     16x16 WMMA tiles, V_WMMA_F32_16X16X4_F32, 64 accumulator VGPRs/wave).
//   - K staged through LDS in chunks of 16 with coalesced b128 global loads;
//     LDS laid out in the WMMA fragment shape (even/odd K interleaved as
//     float2 for A -> ds_load_b64; parity planes for B -> ds_load_b32).
//   - B accessed transposed (Bt[n*ldb+k]) when BT != 0.
//   - blockIdx.z encodes (batch, group): zb = z/Gz, zg = z%Gz.
// Fragment layouts per CDNA5 ISA 7.12.2: A lane lh holds row lh, VGPR0/1 =
// K{0,1} (lanes 0-15) or K{2,3} (lanes 16-31); B/C/D rows striped across
// lanes with the same half-split.
// ---------------------------------------------------------------------------
__global__ __launch_bounds__(128)
void gemm_wmma_f32(const float* __restrict__ A, long lda, long aSB, long aSG,
                   const float* __restrict__ B, long ldb, long bSB, long bSG,
                   int BT,
                   float* __restrict__ C, long ldc, long cSB, long cSG,
                   int K, int Gz, float scale, const float* __restrict__ bias)
{
    __shared__ float2 As2[KC / 2][APAD];     // (k-pair, row) -> {even k, odd k}
    __shared__ float  Bs[2][KC / 2][BPAD];   // (k parity, k-pair, col)

    const int zb = blockIdx.z / Gz;
    const int zg = blockIdx.z % Gz;
    A += (long)zb * aSB + (long)zg * aSG;
    B += (long)zb * bSB + (long)zg * bSG;
    C += (long)zb * cSB + (long)zg * cSG;

    const int tid  = threadIdx.x;
    const int lane = tid & 31;
    const int wave = tid >> 5;
    const int lh   = lane & 15;
    const int hi   = lane >> 4;            // lane half: selects K offset +2

    const int rowBase = blockIdx.y * BM;   // global row of LDS row 0
    const int colBase = blockIdx.x * BN;   // global col of LDS col 0
    const int wcol    = wave * 32;         // wave's col offset in block tile

    v8f acc[4][2] = {};

    for (int k0 = 0; k0 < K; k0 += KC) {
        // ---- stage A chunk: 64 rows x 16 k (2 coalesced float4 per thread)
        #pragma unroll
        for (int i = 0; i < 2; ++i) {
            const int idx = tid + i * 128;          // 0..255
            const int r   = idx >> 2;               // row 0..63
            const int q   = idx & 3;                // float4 within row
            const float4 f =
                *(const float4*)(A + (long)(rowBase + r) * lda + k0 + q * 4);
            As2[2 * q    ][r] = make_float2(f.x, f.y);
            As2[2 * q + 1][r] = make_float2(f.z, f.w);
        }
        // ---- stage B chunk: 16 k x 128 cols (4 coalesced float4 per thread)
        if (BT) {
            #pragma unroll
            for (int i = 0; i < 4; ++i) {
                const int idx = tid + i * 128;      // 0..511
                const int n   = idx >> 2;           // col 0..127
                const int q   = idx & 3;
                const float4 f =
                    *(const float4*)(B + (long)(colBase + n) * ldb + k0 + q * 4);
                Bs[0][2 * q    ][n] = f.x;
                Bs[1][2 * q    ][n] = f.y;
                Bs[0][2 * q + 1][n] = f.z;
                Bs[1][2 * q + 1][n] = f.w;
            }
        } else {
            #pragma unroll
            for (int i = 0; i < 4; ++i) {
                const int idx = tid + i * 128;      // 0..511
                const int kk  = idx >> 5;           // k row 0..15
                const int q   = idx & 31;           // col quad 0..31
                const float4 f =
                    *(const float4*)(B + (long)(k0 + kk) * ldb + colBase + q * 4);
                float* dst = &Bs[kk & 1][kk >> 1][q * 4];
                dst[0] = f.x; dst[1] = f.y; dst[2] = f.z; dst[3] = f.w;
            }
        }
        // prefetch next chunk while we compute this one
        if (k0 + KC < K) {
            __builtin_prefetch(A + (long)(rowBase + (tid >> 1)) * lda + k0 + KC, 0, 0);
            if (BT)
                __builtin_prefetch(B + (long)(colBase + tid) * ldb + k0 + KC, 0, 0);
            else
                __builtin_prefetch(B + (long)(k0 + KC + (tid >> 3)) * ldb + colBase, 0, 0);
        }
        __syncthreads();

        // ---- 4 WMMA K-steps; per step: 4 A frags + 2 B frags -> 8 WMMAs
        #pragma unroll
        for (int ks = 0; ks < 4; ++ks) {
            const int p = 2 * ks + hi;              // K-pair index (kb = 2p)
            v2f a[4], b[2];
            #pragma unroll
            for (int r = 0; r < 4; ++r) {
                const float2 t = As2[p][r * 16 + lh];
                a[r].x = t.x; a[r].y = t.y;
            }
            #pragma unroll
            for (int c = 0; c < 2; ++c) {
                const int col = wcol + c * 16 + lh;
                b[c].x = Bs[0][p][col];
                b[c].y = Bs[1][p][col];
            }
            #pragma unroll
            for (int r = 0; r < 4; ++r)
                #pragma unroll
                for (int c = 0; c < 2; ++c)
                    acc[r][c] = __builtin_amdgcn_wmma_f32_16x16x4_f32(
                        false, a[r], false, b[c], (short)0, acc[r][c],
                        false, false);
        }
        __syncthreads();
    }

    // ---- store: C/D VGPR i holds row (tile + i + 8*hi), col (tile + lh)
    #pragma unroll
    for (int r = 0; r < 4; ++r) {
        #pragma unroll
        for (int c = 0; c < 2; ++c) {
            const int col = colBase + wcol + c * 16 + lh;
            float bv = bias ? bias[col] : 0.f;
            #pragma unroll
            for (int i = 0; i < 8; ++i) {
                const int row = rowBase + r * 16 + i + 8 * hi;
                C[(long)row * ldc + col] = acc[r][c][i] * scale + bv;
            }
        }
    }
}

// ---------------------------------------------------------------------------
// Geometry kernel: per (b, n, m) pair build 64-d sin/cos position embedding,
// dot with WG[16,64], relu, clamp at 1e-6 -> clamped_w[b, n, g, m].
// ---------------------------------------------------------------------------
__global__ __launch_bounds__(256)
void geo_weight(const float* __restrict__ rp, const float* __restrict__ WG,
                float* __restrict__ cw)
{
    __shared__ float sWG[GG * GEO];
    for (int i = threadIdx.x; i < GG * GEO; i += blockDim.x) sWG[i] = WG[i];
    __syncthreads();

    const long idx = (long)blockIdx.x * blockDim.x + threadIdx.x; // b*N*N threads
    const int m = (int)(idx % NN);
    const long t = idx / NN;
    const int n = (int)(t % NN);
    const int b = (int)(t / NN);

    const float* bn = rp + ((long)b * NN + n) * 4;
    const float* bm = rp + ((long)b * NN + m) * 4;
    const float wn = bn[2], hn = bn[3], wm = bm[2], hm = bm[3];
    const float cxn = bn[0] + 0.5f * (wn - 1.f), cyn = bn[1] + 0.5f * (hn - 1.f);
    const float cxm = bm[0] + 0.5f * (wm - 1.f), cym = bm[1] + 0.5f * (hm - 1.f);

    float pos[4];
    pos[0] = __logf(fmaxf(fabsf((cxn - cxm) / wm), 1e-3f));
    pos[1] = __logf(fmaxf(fabsf((cyn - cym) / hm), 1e-3f));
    pos[2] = __logf(wn / wm);
    pos[3] = __logf(hn / hm);

    // dim_mat[j] = 1000^(j/8)
    const float invdim[8] = {1.0f, 1.0f / 2.37137371f, 1.0f / 5.62341325f,
                             1.0f / 13.3352143f, 1.0f / 31.6227766f,
                             1.0f / 74.9894209f, 1.0f / 177.827941f,
                             1.0f / 421.696503f};
    float acc[GG];
    #pragma unroll
    for (int g = 0; g < GG; ++g) acc[g] = 0.f;

    for (int d = 0; d < 4; ++d) {
        for (int j = 0; j < 8; ++j) {
            const float dv = 100.f * pos[d] * invdim[j];
            const float s = __sinf(dv), c = __cosf(dv);
            const int es = d * 16 + j, ec = es + 8; // [sin(8) | cos(8)] per dim
            #pragma unroll
            for (int g = 0; g < GG; ++g)
                acc[g] += sWG[g * GEO + es] * s + sWG[g * GEO + ec] * c;
        }
    }

    float* out = cw + (((long)b * NN + n) * GG) * NN + m;
    #pragma unroll
    for (int g = 0; g < GG; ++g)
        out[(long)g * NN] = fmaxf(acc[g], 1e-6f); // relu then clamp == clamp
}

// ---------------------------------------------------------------------------
// Softmax: one wave32 per row of length 512 (rows = b*N*G, contiguous).
// Writes max-subtracted scores back in place (aff_scale output) and the
// normalized softmax to `smax`.
// ---------------------------------------------------------------------------
__global__ __launch_bounds__(256)
void softmax_rows(float* __restrict__ scores, float* __restrict__ smax, int rows)
{
    const int r = blockIdx.x * (blockDim.x >> 5) + (threadIdx.x >> 5);
    if (r >= rows) return;
    const int lane = threadIdx.x & 31;
    float* rowS = scores + (long)r * NN;
    float* rowO = smax + (long)r * NN;

    float v[16];
    float mx = -3.4e38f;
    #pragma unroll
    for (int j = 0; j < 16; ++j) {
        v[j] = rowS[j * 32 + lane];
        mx = fmaxf(mx, v[j]);
    }
    #pragma unroll
    for (int off = 16; off; off >>= 1) mx = fmaxf(mx, __shfl_xor(mx, off, 32));

    float sum = 0.f;
    #pragma unroll
    for (int j = 0; j < 16; ++j) {
        v[j] -= mx;
        rowS[j * 32 + lane] = v[j];  // aff_scale output (post max-subtract)
        v[j] = __expf(v[j]);
        sum += v[j];
    }
    #pragma unroll
    for (int off = 16; off; off >>= 1) sum += __shfl_xor(sum, off, 32);

    const float inv = 1.f / sum;
    #pragma unroll
    for (int j = 0; j < 16; ++j) rowO[j * 32 + lane] = v[j] * inv;
}

// ---------------------------------------------------------------------------
extern "C" void kernel_launch(void* const* d_in, const int* in_sizes, int n_in,
                              void* d_out, int out_size, void* d_ws, size_t ws_size,
                              hipStream_t stream)
{
    (void)in_sizes; (void)n_in; (void)out_size; (void)ws_size;
    const float* imFtr = (const float*)d_in[0]; // [b, f, p, F] == [b, N, F]
    const float* rp    = (const float*)d_in[1]; // [b, f, p, 4]
    const float* WQ    = (const float*)d_in[2]; // [F, F]
    const float* WK    = (const float*)d_in[3]; // [F, F]
    const float* WG    = (const float*)d_in[4]; // [G, GEO]
    const float* convW = (const float*)d_in[5]; // [F, G*F]
    const float* convB = (const float*)d_in[6]; // [F]

    // Outputs, concatenated flat: lin, aff_softmax, aff_scale, clamped_w
    float* lin    = (float*)d_out;                              //  b*N*F
    float* smax   = lin    + (long)BB * NN * FF;                //  b*N*G*N
    float* ascale = smax   + (long)BB * NN * GG * NN;
    float* cw     = ascale + (long)BB * NN * GG * NN;

    // Workspace: q [b,N,F] | k [b,N,F] | out [b,N,G,F]
    float* qws   = (float*)d_ws;
    float* kws   = qws + (long)BB * NN * FF;
    float* outws = kws + (long)BB * NN * FF;

    const dim3 blk(128); // 4 waves; block tile 64x128

    // 1) geometric weights (independent)
    geo_weight<<<(BB * NN * NN) / 256, 256, 0, stream>>>(rp, WG, cw);

    // 2) q = flat @ WQ^T  (per batch: M=512, N=1024, K=1024)
    gemm_wmma_f32<<<dim3(FF / BN, NN / BM, BB), blk, 0, stream>>>(
        imFtr, FF, (long)NN * FF, 0,
        WQ,    FF, 0, 0, /*BT=*/1,
        qws,   FF, (long)NN * FF, 0,
        FF, 1, 1.0f, nullptr);

    //    k = flat @ WK^T
    gemm_wmma_f32<<<dim3(FF / BN, NN / BM, BB), blk, 0, stream>>>(
        imFtr, FF, (long)NN * FF, 0,
        WK,    FF, 0, 0, /*BT=*/1,
        kws,   FF, (long)NN * FF, 0,
        FF, 1, 1.0f, nullptr);

    // 3) scores[b,n,g,m] = 100 * q_g @ k_g^T  (per (b,g): M=512, N=512, K=64)
    gemm_wmma_f32<<<dim3(NN / BN, NN / BM, BB * GG), blk, 0, stream>>>(
        qws, FF, (long)NN * FF, HD,
        kws, FF, (long)NN * FF, HD, /*BT=*/1,
        ascale, (long)GG * NN, (long)NN * GG * NN, NN,
        HD, GG, 100.0f, nullptr);

    // 4) softmax over m; writes aff_scale (in place) and aff_softmax
    softmax_rows<<<(BB * NN * GG) / 8, 256, 0, stream>>>(ascale, smax, BB * NN * GG);

    // 5) out[b,n,g,:] = softmax_g @ flat  (per (b,g): M=512, N=1024, K=512)
    gemm_wmma_f32<<<dim3(FF / BN, NN / BM, BB * GG), blk, 0, stream>>>(
        smax,  (long)GG * NN, (long)NN * GG * NN, NN,
        imFtr, FF,            (long)NN * FF, 0, /*BT=*/0,
        outws, (long)GG * FF, (long)NN * GG * FF, FF,
        NN, GG, 1.0f, nullptr);

    // 6) lin = out2d[1024,16384] @ conv_w^T + conv_b  (M=1024, N=1024, K=16384)
    gemm_wmma_f32<<<dim3(FF / BN, (BB * NN) / BM, 1), blk, 0, stream>>>(
        outws, (long)GG * FF, 0, 0,
        convW, (long)GG * FF, 0, 0, /*BT=*/1,
        lin,   FF, 0, 0,
        GG * FF, 1, 1.0f, convB);
}